// IP_Basic_71923522339376
// MI455X (gfx1250) — compile-verified
//
#include <hip/hip_runtime.h>
#include <stdint.h>

// ---------------------------------------------------------------------------
// Fused IP-Basic depth completion for MI455X (gfx1250, wave32).
// One kernel, one HBM pass; all intermediates in LDS (320KB/WGP).
//   invert -> dilate5 -> dilate5 -> erode5 -> fill(dilate7) -> median5
//          -> gauss5(reflect) -> invert
// CDNA5 paths: GLOBAL_LOAD_ASYNC_TO_LDS_B128 tile staging (ASYNCcnt),
// s_wait_asynccnt, wave32 blocks. Interior tiles (73%) run a check-free path.
// ---------------------------------------------------------------------------

#define H_IMG 352
#define W_IMG 1216
#define BATCH 16

#define TH 32
#define TW 64
#define YH 13                 // staged y halo (total stencil radius)
#define XH 16                 // staged x halo -> 16B-aligned b128 async loads
#define RH (TH + 2 * YH)      // 58 staged rows
#define RWS (TW + 2 * XH)     // 96 staged cols
#define PITCH RWS             // rows 16B-aligned in LDS
#define NCH (RWS / 4)         // 24 b128 chunks per staged row
#define NT 256                // 8 wave32 waves

#define MIN_D 0.1f
#define MAX_D 100.0f
#define NEG_BIG (-3.402823466e38f)
#define POS_BIG ( 3.402823466e38f)

// ---- CDNA5 async global->LDS (LDS byte offset = low 32 bits of generic ptr)
__device__ __forceinline__ void async_ld_b128(uint32_t lds_off, const float* g) {
    asm volatile("global_load_async_to_lds_b128 %0, %1, off"
                 :: "v"(lds_off), "v"(g) : "memory");
}
__device__ __forceinline__ void wait_async0() {
    asm volatile("s_wait_asynccnt 0" ::: "memory");
}

// ---- exact median-of-25: Devillard opt_med25, 99 compare-exchanges --------
#define CEx(a, b) { float _t = fminf(a, b); b = fmaxf(a, b); a = _t; }
__device__ __forceinline__ float med25(float p[25]) {
    CEx(p[0],p[1]);  CEx(p[3],p[4]);  CEx(p[2],p[4]);
    CEx(p[2],p[3]);  CEx(p[6],p[7]);  CEx(p[5],p[7]);
    CEx(p[5],p[6]);  CEx(p[9],p[10]); CEx(p[8],p[10]);
    CEx(p[8],p[9]);  CEx(p[12],p[13]);CEx(p[11],p[13]);
    CEx(p[11],p[12]);CEx(p[15],p[16]);CEx(p[14],p[16]);
    CEx(p[14],p[15]);CEx(p[18],p[19]);CEx(p[17],p[19]);
    CEx(p[17],p[18]);CEx(p[21],p[22]);CEx(p[20],p[22]);
    CEx(p[20],p[21]);CEx(p[23],p[24]);CEx(p[2],p[5]);
    CEx(p[3],p[6]);  CEx(p[0],p[6]);  CEx(p[0],p[3]);
    CEx(p[4],p[7]);  CEx(p[1],p[7]);  CEx(p[1],p[4]);
    CEx(p[11],p[14]);CEx(p[8],p[14]); CEx(p[8],p[11]);
    CEx(p[12],p[15]);CEx(p[9],p[15]); CEx(p[9],p[12]);
    CEx(p[13],p[16]);CEx(p[10],p[16]);CEx(p[10],p[13]);
    CEx(p[20],p[23]);CEx(p[17],p[23]);CEx(p[17],p[20]);
    CEx(p[21],p[24]);CEx(p[18],p[24]);CEx(p[18],p[21]);
    CEx(p[19],p[22]);CEx(p[8],p[17]); CEx(p[9],p[18]);
    CEx(p[0],p[18]); CEx(p[0],p[9]);  CEx(p[10],p[19]);
    CEx(p[1],p[19]); CEx(p[1],p[10]); CEx(p[11],p[20]);
    CEx(p[2],p[20]); CEx(p[2],p[11]); CEx(p[12],p[21]);
    CEx(p[3],p[21]); CEx(p[3],p[12]); CEx(p[13],p[22]);
    CEx(p[4],p[22]); CEx(p[4],p[13]); CEx(p[14],p[23]);
    CEx(p[5],p[23]); CEx(p[5],p[14]); CEx(p[15],p[24]);
    CEx(p[6],p[24]); CEx(p[6],p[15]); CEx(p[7],p[16]);
    CEx(p[7],p[19]); CEx(p[13],p[21]);CEx(p[15],p[23]);
    CEx(p[7],p[13]); CEx(p[7],p[15]); CEx(p[1],p[9]);
    CEx(p[3],p[11]); CEx(p[5],p[17]); CEx(p[11],p[17]);
    CEx(p[9],p[17]); CEx(p[4],p[10]); CEx(p[6],p[12]);
    CEx(p[7],p[14]); CEx(p[4],p[6]);  CEx(p[4],p[7]);
    CEx(p[12],p[14]);CEx(p[10],p[14]);CEx(p[6],p[7]);
    CEx(p[10],p[12]);CEx(p[6],p[10]); CEx(p[6],p[17]);
    CEx(p[12],p[17]);CEx(p[7],p[17]); CEx(p[7],p[10]);
    CEx(p[12],p[18]);CEx(p[7],p[12]); CEx(p[10],p[18]);
    CEx(p[12],p[20]);CEx(p[10],p[20]);CEx(p[10],p[12]);
    return p[12];
}

// ---- horizontal dilate5 with fused input inversion (first stage) ----------
template <bool EDGE>
__device__ __forceinline__ void hpass_inv(const float* __restrict__ src,
                                          float* __restrict__ dst,
                                          int tx0, int tid) {
    const int hy = 13, hx = 11;
    const int RX = TW + 2 * hx, RY = TH + 2 * hy;
    for (int i = tid; i < RY * RX; i += NT) {
        const int ly = i / RX + (YH - hy);
        const int lx = i % RX + (XH - hx);
        const int gxc = tx0 + lx - XH;
        float m = NEG_BIG;
#pragma unroll
        for (int d = -2; d <= 2; ++d) {
            const float raw = src[ly * PITCH + lx + d];
            float v = (raw > MIN_D) ? (MAX_D - raw) : raw;   // fused invert
            if (EDGE) {
                const int gx = gxc + d;
                v = (gx >= 0 && gx < W_IMG) ? v : NEG_BIG;
            }
            m = fmaxf(m, v);
        }
        dst[ly * PITCH + lx] = m;
    }
}

template <int R, bool IS_MIN, bool EDGE>
__device__ __forceinline__ void hpass(const float* __restrict__ src,
                                      float* __restrict__ dst,
                                      int hy, int hx, int tx0, int tid) {
    const int RX = TW + 2 * hx, RY = TH + 2 * hy;
    const float PV = IS_MIN ? POS_BIG : NEG_BIG;
    for (int i = tid; i < RY * RX; i += NT) {
        const int ly = i / RX + (YH - hy);
        const int lx = i % RX + (XH - hx);
        const int gxc = tx0 + lx - XH;
        float m = PV;
#pragma unroll
        for (int d = -R; d <= R; ++d) {
            float v = src[ly * PITCH + lx + d];
            if (EDGE) {
                const int gx = gxc + d;
                v = (gx >= 0 && gx < W_IMG) ? v : PV;
            }
            m = IS_MIN ? fminf(m, v) : fmaxf(m, v);
        }
        dst[ly * PITCH + lx] = m;
    }
}

template <int R, bool IS_MIN, bool EDGE>
__device__ __forceinline__ void vpass(const float* __restrict__ src,
                                      float* __restrict__ dst,
                                      int hy, int hx, int ty0, int tid) {
    const int RX = TW + 2 * hx, RY = TH + 2 * hy;
    const float PV = IS_MIN ? POS_BIG : NEG_BIG;
    for (int i = tid; i < RY * RX; i += NT) {
        const int ly = i / RX + (YH - hy);
        const int lx = i % RX + (XH - hx);
        const int gyc = ty0 + ly - YH;
        float m = PV;
#pragma unroll
        for (int d = -R; d <= R; ++d) {
            float v = src[(ly + d) * PITCH + lx];
            if (EDGE) {
                const int gy = gyc + d;
                v = (gy >= 0 && gy < H_IMG) ? v : PV;
            }
            m = IS_MIN ? fminf(m, v) : fmaxf(m, v);
        }
        dst[ly * PITCH + lx] = m;
    }
}

template <bool EDGE>
__device__ __forceinline__ void pipeline(float* __restrict__ A,
                                         float* __restrict__ Bf,
                                         const float* __restrict__ in,
                                         float* __restrict__ out,
                                         int tx0, int ty0, int b, int tid) {
    const float* img = in + (size_t)b * H_IMG * W_IMG;

    // ---- Stage 0: async b128 staging of 58x96 region into A ----------------
    for (int i = tid; i < RH * NCH; i += NT) {
        const int ly = i / NCH, c = i % NCH;
        const int gy = ty0 + ly - YH;
        const int gx = tx0 - XH + 4 * c;         // 16B aligned; never straddles bounds
        bool ok = true;
        if (EDGE) ok = (gy >= 0 && gy < H_IMG && gx >= 0 && gx + 3 < W_IMG);
        if (ok) {
            const uint32_t lds = (uint32_t)(uintptr_t)(&A[ly * PITCH + 4 * c]);
            async_ld_b128(lds, img + (size_t)gy * W_IMG + gx);
        }
    }
    wait_async0();      // each wave drains its own ASYNCcnt ...
    __syncthreads();    // ... then barrier -> all waves' LDS writes visible

    // ---- Stage 1: dilate5 (invert fused into h-pass), out halo 11 ----------
    hpass_inv<EDGE>(A, Bf, tx0, tid);                 __syncthreads();
    vpass<2, false, EDGE>(Bf, A, 11, 11, ty0, tid);   __syncthreads();

    // ---- Stage 2: closing = dilate5 then erode5 ----------------------------
    hpass<2, false, EDGE>(A, Bf, 11, 9, tx0, tid);    __syncthreads();
    vpass<2, false, EDGE>(Bf, A, 9, 9, ty0, tid);     __syncthreads();  // dilate, halo 9
    hpass<2, true,  EDGE>(A, Bf, 9, 7, tx0, tid);     __syncthreads();
    vpass<2, true,  EDGE>(Bf, A, 7, 7, ty0, tid);     __syncthreads();  // erode, halo 7

    // ---- Stage 3: fill empty with dilate7 (separable), out halo 4 ----------
    hpass<3, false, EDGE>(A, Bf, 7, 4, tx0, tid);     __syncthreads();
    {
        const int RX = TW + 8, RY = TH + 8;
        for (int i = tid; i < RY * RX; i += NT) {
            const int ly = i / RX + (YH - 4);
            const int lx = i % RX + (XH - 4);
            const int gyc = ty0 + ly - YH;
            float m = NEG_BIG;
#pragma unroll
            for (int d = -3; d <= 3; ++d) {
                float v = Bf[(ly + d) * PITCH + lx];
                if (EDGE) {
                    const int gy = gyc + d;
                    v = (gy >= 0 && gy < H_IMG) ? v : NEG_BIG;
                }
                m = fmaxf(m, v);
            }
            const float orig = A[ly * PITCH + lx];
            A[ly * PITCH + lx] = (orig < MIN_D) ? m : orig;
        }
    }
    __syncthreads();

    // ---- Stage 4: 5x5 exact median (zero padding), out halo 2 --------------
    {
        const int RX = TW + 4, RY = TH + 4;
        for (int i = tid; i < RY * RX; i += NT) {
            const int ly = i / RX + (YH - 2);
            const int lx = i % RX + (XH - 2);
            const int gyc = ty0 + ly - YH, gxc = tx0 + lx - XH;
            float p[25];
            int n = 0;
#pragma unroll
            for (int dy = -2; dy <= 2; ++dy)
#pragma unroll
                for (int dx = -2; dx <= 2; ++dx) {
                    float v = A[(ly + dy) * PITCH + lx + dx];
                    if (EDGE) {
                        const int gy = gyc + dy, gx = gxc + dx;
                        v = (gy >= 0 && gy < H_IMG && gx >= 0 && gx < W_IMG) ? v : 0.0f;
                    }
                    p[n++] = v;
                }
            Bf[ly * PITCH + lx] = med25(p);
        }
    }
    __syncthreads();

    // ---- Stage 5: separable 5-tap Gaussian (sigma=1.1, reflect) + finish ---
    float w_[5];
    {
        const float e1 = expf(-1.0f / (2.0f * 1.1f * 1.1f));
        const float e2 = expf(-4.0f / (2.0f * 1.1f * 1.1f));
        const float s  = 1.0f + 2.0f * e1 + 2.0f * e2;
        w_[0] = e2 / s; w_[1] = e1 / s; w_[2] = 1.0f / s; w_[3] = e1 / s; w_[4] = e2 / s;
    }
    {   // horizontal: Bf -> A, rows halo 2, cols halo 0
        const int RX = TW, RY = TH + 4;
        for (int i = tid; i < RY * RX; i += NT) {
            const int ly = i / RX + (YH - 2);
            const int lx = i % RX + XH;
            float s = 0.0f;
            if (EDGE) {
                const int gy = ty0 + ly - YH;
                if (gy < 0 || gy >= H_IMG) continue;   // never consumed after reflect
                const int gxc = tx0 + lx - XH;
#pragma unroll
                for (int d = -2; d <= 2; ++d) {
                    int gx = gxc + d;
                    gx = (gx < 0) ? -gx : gx;                       // reflect, no edge repeat
                    gx = (gx >= W_IMG) ? (2 * W_IMG - 2 - gx) : gx;
                    s += w_[d + 2] * Bf[ly * PITCH + (gx - tx0 + XH)];
                }
            } else {
#pragma unroll
                for (int d = -2; d <= 2; ++d)
                    s += w_[d + 2] * Bf[ly * PITCH + lx + d];
            }
            A[ly * PITCH + lx] = s;
        }
    }
    __syncthreads();
    {   // vertical + valid-mask merge + inverse transform + store
        for (int i = tid; i < TH * TW; i += NT) {
            const int ly = i / TW + YH, lx = i % TW + XH;
            const int gy0 = ty0 + ly - YH, gx0 = tx0 + lx - XH;
            float s = 0.0f;
            if (EDGE) {
#pragma unroll
                for (int d = -2; d <= 2; ++d) {
                    int gy = gy0 + d;
                    gy = (gy < 0) ? -gy : gy;
                    gy = (gy >= H_IMG) ? (2 * H_IMG - 2 - gy) : gy;
                    s += w_[d + 2] * A[(gy - ty0 + YH) * PITCH + lx];
                }
            } else {
#pragma unroll
                for (int d = -2; d <= 2; ++d)
                    s += w_[d + 2] * A[(ly + d) * PITCH + lx];
            }
            const float dm = Bf[ly * PITCH + lx];
            float v = (dm > MIN_D) ? s : dm;      // blur only valid pixels
            v = (v > MIN_D) ? (MAX_D - v) : v;    // invert back
            out[((size_t)b * H_IMG + gy0) * W_IMG + gx0] = v;
        }
    }
}

extern "C" __global__ void __launch_bounds__(NT)
ip_basic_fused(const float* __restrict__ in, float* __restrict__ out) {
    __shared__ __align__(16) float A[RH * PITCH];
    __shared__ __align__(16) float Bf[RH * PITCH];

    const int tid = threadIdx.x;
    const int tx0 = blockIdx.x * TW;
    const int ty0 = blockIdx.y * TH;
    const int b   = blockIdx.z;

    // block-uniform branch: border ring needs padding semantics, interior doesn't
    const bool edge = (blockIdx.x == 0) || (blockIdx.x == (W_IMG / TW - 1)) ||
                      (blockIdx.y == 0) || (blockIdx.y == (H_IMG / TH - 1));
    if (edge) pipeline<true >(A, Bf, in, out, tx0, ty0, b, tid);
    else      pipeline<false>(A, Bf, in, out, tx0, ty0, b, tid);
}

extern "C" void kernel_launch(void* const* d_in, const int* in_sizes, int n_in,
                              void* d_out, int out_size, void* d_ws, size_t ws_size,
                              hipStream_t stream) {
    (void)in_sizes; (void)n_in; (void)out_size; (void)d_ws; (void)ws_size;
    const float* in = (const float*)d_in[0];
    float* out = (float*)d_out;
    dim3 grid(W_IMG / TW, H_IMG / TH, BATCH);   // 19 x 11 x 16, exact tiling
    ip_basic_fused<<<grid, dim3(NT), 0, stream>>>(in, out);
}